// X_73813307949716
// MI455X (gfx1250) — compile-verified
//
#include <hip/hip_runtime.h>
#include <cstdint>

// Reference collapses to a row permutation: out[j,b] = x[j^2048, b] per plane
// (Pauli-X on qubit 0 = flip bit 11 of the row index). Flat form per plane:
// out[e] = in[e ^ (1<<18)]. Aligned chunks below the bit-18 boundary map to
// contiguous source chunks, so the whole op is 2*2 contiguous 1 MiB copies.
//
// Pure data movement (8 MiB total, ~0.36us roofline at 23.3 TB/s). The CDNA5
// path for bulk copies is the Tensor Data Mover: one tensor_load_to_lds +
// one tensor_store_from_lds per 16 KiB chunk, tracked by TENSORcnt, with the
// descriptor (D#) built in SGPRs per ISA ch.8.

typedef uint32_t u32x4 __attribute__((ext_vector_type(4)));
typedef uint32_t u32x8 __attribute__((ext_vector_type(8)));

#define PLANE_FLOATS  (4096 * 128)            // 524288 floats per (real|imag) plane
#define XOR_FLOATS    (2048 * 128)            // 262144: flip qubit-0 row bit (bit 18 flat)
#define CHUNK_FLOATS  4096                    // 16 KiB per TDM transfer
#define N_TOTAL       (2 * PLANE_FLOATS)      // 1048576 output floats
#define NBLOCKS       (N_TOTAL / CHUNK_FLOATS) // 256 workgroups

// D# group 0: [1:0]=count, [63:32]=lds_addr, [120:64]=global_addr, [127:126]=type(2)
__device__ __forceinline__ u32x4 make_g0(uint64_t gaddr, uint32_t lds_addr) {
    u32x4 g0;
    g0.x = 1u;                                                    // count = 1 (valid)
    g0.y = lds_addr;                                              // LDS byte address
    g0.z = (uint32_t)(gaddr & 0xffffffffu);                       // global_addr[31:0]
    g0.w = (uint32_t)((gaddr >> 32) & 0x1ffffffu) | 0x80000000u;  // addr[56:32] | type=2
    return g0;
}

__global__ __launch_bounds__(32)
void qushift_tdm(const float* __restrict__ xr,
                 const float* __restrict__ xi,
                 float* __restrict__ out)
{
    __shared__ float smem[CHUNK_FLOATS];      // 16 KiB staging tile

    // Per-workgroup chunk of the flat [2*D*B] output.
    const uint32_t e0    = blockIdx.x * CHUNK_FLOATS;
    const uint32_t plane = e0 >> 19;                  // 0 = real, 1 = imag
    const uint32_t p     = e0 & (PLANE_FLOATS - 1);   // offset within plane
    const float* srcp = (plane ? xi : xr) + (p ^ XOR_FLOATS);
    float*       dstp = out + e0;

    const uint32_t lds = (uint32_t)(uintptr_t)&smem[0];

    // D# group 1: data_size=4B; 1-D tile of CHUNK_FLOATS elements.
    // [17:16]=data_size(2); [79:48]=tensor_dim0; [111:80]=tensor_dim1;
    // [127:112]=tile_dim0; [143:128]=tile_dim1; [207:160]=tensor_dim0_stride.
    u32x8 g1 = {0, 0, 0, 0, 0, 0, 0, 0};
    g1.s0 = (2u << 16);                           // data_size = 4 bytes
    g1.s1 = ((uint32_t)CHUNK_FLOATS << 16);       // tensor_dim0[15:0]
    g1.s2 = (1u << 16);                           // tensor_dim0[31:16]=0 | tensor_dim1[15:0]=1
    g1.s3 = ((uint32_t)CHUNK_FLOATS << 16);       // tensor_dim1[31:16]=0 | tile_dim0
    g1.s4 = 1u;                                   // tile_dim1=1, tile_dim2=0
    g1.s5 = (uint32_t)CHUNK_FLOATS;               // tensor_dim0_stride[31:0]

    const u32x4 z4 = {0, 0, 0, 0};                // groups 2/3 unused (<=2-D tensor)
    const u32x4 g0_load  = make_g0((uint64_t)(uintptr_t)srcp, lds);
    const u32x4 g0_store = make_g0((uint64_t)(uintptr_t)dstp, lds);

    // Global -> LDS DMA (TENSORcnt++), wait, LDS -> Global DMA, wait.
    asm volatile("tensor_load_to_lds %0, %1, %2, %3"
                 :: "s"(g0_load), "s"(g1), "s"(z4), "s"(z4)
                 : "memory");
    __builtin_amdgcn_s_wait_tensorcnt(0);
    asm volatile("tensor_store_from_lds %0, %1, %2, %3"
                 :: "s"(g0_store), "s"(g1), "s"(z4), "s"(z4)
                 : "memory");
    __builtin_amdgcn_s_wait_tensorcnt(0);
}

extern "C" void kernel_launch(void* const* d_in, const int* in_sizes, int n_in,
                              void* d_out, int out_size, void* d_ws, size_t ws_size,
                              hipStream_t stream)
{
    const float* xr  = (const float*)d_in[0];
    const float* xi  = (const float*)d_in[1];
    float*       out = (float*)d_out;

    // One wave per block: exactly one TDM load+store pair per 16 KiB chunk.
    qushift_tdm<<<dim3(NBLOCKS), dim3(32), 0, stream>>>(xr, xi, out);
}